// Attention_54580444397738
// MI455X (gfx1250) — compile-verified
//
#include <hip/hip_runtime.h>
#include <math.h>

// CDNA5 wave32 WMMA / vector types (clang native vectors, builtin-compatible)
typedef __attribute__((ext_vector_type(2))) float v2f;
typedef __attribute__((ext_vector_type(4))) float v4f;
typedef __attribute__((ext_vector_type(8))) float v8f;

#define Bq   8
#define Nq   128
#define Hq   256
#define BINq 16

static __device__ __forceinline__ v8f wmma_f32_4(v2f a, v2f b, v8f c) {
    // D = A(16x4,f32) * B(4x16,f32) + C(16x16,f32)   -> v_wmma_f32_16x16x4_f32
    return __builtin_amdgcn_wmma_f32_16x16x4_f32(false, a, false, b, (short)0, c,
                                                 false, false);
}

// ---------------------------------------------------------------------------
// Kernel 1: U[b,n,h] = local[b,n,:] @ W_apair[:,h] + 0.5*(b_apair[h]+b_binary[h])
// M = B*N = 1024 rows, K = 256, N = 256.  One wave per 16x16 tile.
// ---------------------------------------------------------------------------
__global__ void __launch_bounds__(128) compute_U_kernel(
    const float* __restrict__ local, const float* __restrict__ Wap,
    const float* __restrict__ bap, const float* __restrict__ bbin,
    float* __restrict__ U) {
    const int lane = threadIdx.x & 31;
    const int wave = threadIdx.x >> 5;
    const int tile = blockIdx.x * 4 + wave;     // 64 * 16 = 1024 tiles
    const int tM   = tile >> 4;                 // row tile (B*N/16)
    const int tN   = tile & 15;                 // col tile (H/16)
    const int half = lane >> 4;                 // 0: lanes 0-15, 1: lanes 16-31
    const int l15  = lane & 15;

    const int rowA = tM * 16 + l15;             // A-matrix row (M), both halves
    const int colB = tN * 16 + l15;             // B-matrix col (N = h)

    v8f acc = {};
    #pragma unroll 4
    for (int k = 0; k < Hq; k += 4) {
        const int ka = k + half * 2;            // VGPR0 holds K=ka, VGPR1 K=ka+1
        v2f a; a.x = local[rowA * Hq + ka];  a.y = local[rowA * Hq + ka + 1];
        v2f b; b.x = Wap[ka * Hq + colB];    b.y = Wap[(ka + 1) * Hq + colB];
        acc = wmma_f32_4(a, b, acc);
    }
    const float bias = 0.5f * (bap[colB] + bbin[colB]);
    const int rowD = tM * 16 + half * 8;        // D: VGPR r -> rows r / r+8
    #pragma unroll
    for (int r = 0; r < 8; ++r)
        U[(size_t)(rowD + r) * Hq + colB] = acc[r] + bias;
}

// ---------------------------------------------------------------------------
// Kernel 2: per tile of 16 consecutive pairs p=(b,i,j0..j0+15):
//   V(16x256) = binary(16x16) @ W_binary(16x256)        (WMMA, K=16)
//   t = relu(U_i + U_j + V);  score = sigmoid(t . W_att + b_att)
// binary_feats (8.4 MB) is streamed exactly once -> non-temporal loads so it
// never evicts the hot W_binary / U / W_att tables from WGP$/L2.
// ---------------------------------------------------------------------------
__global__ void __launch_bounds__(128) pair_score_kernel(
    const float* __restrict__ bin,   // [B,N,N,16]
    const float* __restrict__ Wb,    // [16,256]
    const float* __restrict__ U,     // [B*N,256]
    const float* __restrict__ Watt,  // [256]
    const float* __restrict__ batt,  // [1]
    float* __restrict__ score) {     // [B*N*N]
    const int lane = threadIdx.x & 31;
    const int wave = threadIdx.x >> 5;
    const int tile = blockIdx.x * 4 + wave;     // 8192 tiles of 16 pairs
    const int p0   = tile * 16;
    const int b    = p0 >> 14;                  // / (N*N)
    const int i    = (p0 >> 7) & (Nq - 1);
    const int j0   = p0 & (Nq - 1);
    const int half = lane >> 4;
    const int l15  = lane & 15;

    // Preload the 16x16 A block (binary rows) for all 4 K-steps (NT stream).
    v2f a[4];
    const float* binRow = bin + (size_t)(p0 + l15) * BINq;
    #pragma unroll
    for (int kk = 0; kk < 4; ++kk) {
        const int ka = kk * 4 + half * 2;
        a[kk].x = __builtin_nontemporal_load(binRow + ka);
        a[kk].y = __builtin_nontemporal_load(binRow + ka + 1);
    }

    const float* Ui  = U + (size_t)(b * Nq + i)  * Hq;
    const float* Uj0 = U + (size_t)(b * Nq + j0) * Hq;

    float partial[8] = {0.f, 0.f, 0.f, 0.f, 0.f, 0.f, 0.f, 0.f};

    for (int hc = 0; hc < Hq / 16; ++hc) {
        const int h = hc * 16 + l15;            // this lane's output column
        v8f acc = {};
        #pragma unroll
        for (int kk = 0; kk < 4; ++kk) {
            const int ka = kk * 4 + half * 2;
            v2f bm; bm.x = Wb[ka * Hq + h]; bm.y = Wb[(ka + 1) * Hq + h];
            acc = wmma_f32_4(a[kk], bm, acc);
        }
        const float ui = Ui[h];
        const float wa = Watt[h];
        #pragma unroll
        for (int r = 0; r < 8; ++r) {           // row m = r + half*8, j = j0+m
            const int m = r + half * 8;
            float t = ui + Uj0[(size_t)m * Hq + h] + acc[r];
            t = t > 0.f ? t : 0.f;
            partial[r] += t * wa;
        }
    }
    // Butterfly-reduce across the 16 lanes of each half (masks stay in-half).
    #pragma unroll
    for (int r = 0; r < 8; ++r) {
        float v = partial[r];
        v += __shfl_xor(v, 1, 32);
        v += __shfl_xor(v, 2, 32);
        v += __shfl_xor(v, 4, 32);
        v += __shfl_xor(v, 8, 32);
        partial[r] = v;
    }
    if (l15 == 0) {
        const float ba = batt[0];
        #pragma unroll
        for (int r = 0; r < 8; ++r) {
            const float s = 1.f / (1.f + __expf(-(partial[r] + ba)));
            score[p0 + half * 8 + r] = s;
        }
    }
}

// ---------------------------------------------------------------------------
// Kernel 3: G[b,i,h] = sum_j score[b,i,j] * local[b,j,h]
// Per batch: 128x256 = score(128x128) @ local(128x256). One wave per tile.
// ---------------------------------------------------------------------------
__global__ void __launch_bounds__(128) global_feat_kernel(
    const float* __restrict__ score, // [B,N,N]
    const float* __restrict__ local, // [B,N,H]
    float* __restrict__ G) {         // [B,N,H]
    const int lane = threadIdx.x & 31;
    const int wave = threadIdx.x >> 5;
    const int tile = blockIdx.x * 4 + wave;     // 8 * 8 * 16 = 1024 tiles
    const int b    = tile >> 7;
    const int tI   = (tile >> 4) & 7;
    const int tH   = tile & 15;
    const int half = lane >> 4;
    const int l15  = lane & 15;

    const float* S = score + (size_t)b * Nq * Nq;
    const float* L = local + (size_t)b * Nq * Hq;
    const int rowA = tI * 16 + l15;
    const int colB = tH * 16 + l15;

    v8f acc = {};
    #pragma unroll 4
    for (int k = 0; k < Nq; k += 4) {
        const int ka = k + half * 2;
        v2f a;  a.x  = S[rowA * Nq + ka];    a.y  = S[rowA * Nq + ka + 1];
        v2f bm; bm.x = L[ka * Hq + colB];    bm.y = L[(ka + 1) * Hq + colB];
        acc = wmma_f32_4(a, bm, acc);
    }
    const int rowD = tI * 16 + half * 8;
    #pragma unroll
    for (int r = 0; r < 8; ++r)
        G[((size_t)b * Nq + rowD + r) * Hq + colB] = acc[r];
}

// ---------------------------------------------------------------------------
// Kernel 4: the bandwidth roofline — write both 134 MB outputs.
//   local_pair[b,i,j,:] = local[b,i,:] + local[b,j,:]
//   global_pair[b,i,j,:] = G[b,i,:] + G[b,j,:]
// 268 MB of stores > 192 MB L2 and never re-read -> non-temporal b128 stores
// so they don't evict the 1 MB local/G tables this kernel re-reads ~128x each.
// ---------------------------------------------------------------------------
__global__ void __launch_bounds__(256) pair_write_kernel(
    const float* __restrict__ local, const float* __restrict__ G,
    float* __restrict__ lp, float* __restrict__ gp) {
    const size_t idx = (size_t)blockIdx.x * blockDim.x + threadIdx.x; // B*N*N*H/4
    const int    hv  = (int)(idx & (Hq / 4 - 1));
    const size_t pij = idx >> 6;
    const int j = (int)(pij & (Nq - 1));
    const int i = (int)((pij >> 7) & (Nq - 1));
    const int b = (int)(pij >> 14);

    const v4f* Li = (const v4f*)(local + (size_t)(b * Nq + i) * Hq) + hv;
    const v4f* Lj = (const v4f*)(local + (size_t)(b * Nq + j) * Hq) + hv;
    const v4f* Gi = (const v4f*)(G     + (size_t)(b * Nq + i) * Hq) + hv;
    const v4f* Gj = (const v4f*)(G     + (size_t)(b * Nq + j) * Hq) + hv;

    const v4f o1 = *Li + *Lj;
    const v4f o2 = *Gi + *Gj;
    __builtin_nontemporal_store(o1, ((v4f*)lp) + idx);
    __builtin_nontemporal_store(o2, ((v4f*)gp) + idx);
}

// ---------------------------------------------------------------------------
extern "C" void kernel_launch(void* const* d_in, const int* in_sizes, int n_in,
                              void* d_out, int out_size, void* d_ws, size_t ws_size,
                              hipStream_t stream) {
    const float* local = (const float*)d_in[0];  // [B,N,H]
    const float* bin   = (const float*)d_in[1];  // [B,N,N,BIN]
    const float* Wap   = (const float*)d_in[2];  // [H,H]
    const float* bap   = (const float*)d_in[3];  // [H]
    const float* Wb    = (const float*)d_in[4];  // [BIN,H]
    const float* bbin  = (const float*)d_in[5];  // [H]
    const float* Watt  = (const float*)d_in[6];  // [H,1]
    const float* batt  = (const float*)d_in[7];  // [1]

    const size_t pair_elems = (size_t)Bq * Nq * Nq * Hq;   // 33,554,432
    float* lp = (float*)d_out;                   // local_pair
    float* gp = (float*)d_out + pair_elems;      // global_pair

    // Workspace carve-out: U (1MB) | G (1MB) | score (0.5MB)
    char*  ws    = (char*)d_ws;
    float* U     = (float*)ws;
    float* G     = (float*)(ws + (size_t)Bq * Nq * Hq * sizeof(float));
    float* score = (float*)(ws + (size_t)2 * Bq * Nq * Hq * sizeof(float));

    // 1) U = local @ W_apair + 0.5*(b_apair+b_binary)     (1024 wave-tiles)
    compute_U_kernel<<<256, 128, 0, stream>>>(local, Wap, bap, bbin, U);
    // 2) score for all B*N*N pairs                        (8192 wave-tiles)
    pair_score_kernel<<<2048, 128, 0, stream>>>(bin, Wb, U, Watt, batt, score);
    // 3) G = score @ local per batch                      (1024 wave-tiles)
    global_feat_kernel<<<256, 128, 0, stream>>>(score, local, G);
    // 4) stream out both pair tensors (268 MB, the roofline)
    const size_t vec4 = pair_elems / 4;          // 8,388,608 threads
    pair_write_kernel<<<(unsigned)(vec4 / 256), 256, 0, stream>>>(local, G, lp, gp);
}